// GCRNN_12704513262021
// MI455X (gfx1250) — compile-verified
//
#include <hip/hip_runtime.h>
#include <hip/hip_bf16.h>
#include <math.h>
#include <stdint.h>

typedef float v2f __attribute__((ext_vector_type(2)));
typedef float v8f __attribute__((ext_vector_type(8)));

#define NN 50000
#define NE 800000
#define NG 64

// ---------------------------------------------------------------------------
// CDNA5 async global->LDS staging (ASYNCcnt-tracked, bypasses VGPRs).
// Flat shared-pointer low 32 bits == wave-relative LDS byte offset (ISA 10.2).
// ---------------------------------------------------------------------------

__device__ __forceinline__ void async_ld_b128(const void* g, void* l) {
    unsigned lds = (unsigned)(uintptr_t)l;
    asm volatile("global_load_async_to_lds_b128 %0, %1, off"
                 :: "v"(lds), "v"(g) : "memory");
}

__device__ __forceinline__ void wait_async0() {
    asm volatile("s_wait_asynccnt 0x0" ::: "memory");
}

// ---------------------------------------------------------------------------
// Graph / elementwise kernels
// ---------------------------------------------------------------------------

__global__ void k_init(float* deg, float* pooled, float* cnt, int n) {
    int i = blockIdx.x * blockDim.x + threadIdx.x;
    if (i < n) deg[i] = 1.0f;                // self-loop contributes 1
    if (i < NG * 128) pooled[i] = 0.0f;
    if (i < NG) cnt[i] = 0.0f;
}

__global__ void k_degree(const int* __restrict__ ei, float* __restrict__ deg, int E) {
    int e = blockIdx.x * blockDim.x + threadIdx.x;
    if (e < E) atomicAdd(&deg[ei[E + e]], 1.0f);   // dst row
}

__global__ void k_rsqrt(float* d, int n) {
    int i = blockIdx.x * blockDim.x + threadIdx.x;
    if (i < n) d[i] = rsqrtf(d[i]);
}

// agg[i] = feat[i] * dinv[i]^2   (self-loop message, also initializes agg)
template<int D>
__global__ void k_self(const float* __restrict__ feat, const float* __restrict__ dinv,
                       float* __restrict__ agg, int n) {
    long j = (long)blockIdx.x * blockDim.x + threadIdx.x;
    if (j >= (long)n * D) return;
    int i = (int)(j / D);
    float di = dinv[i];
    agg[j] = feat[j] * di * di;
}

// agg[dst] += feat[src] * dinv[src]*dinv[dst]
// 128 consecutive threads share one edge -> src/dst/dinv loads are wave-uniform.
template<int D>
__global__ void k_edge_agg(const float* __restrict__ feat, const int* __restrict__ ei,
                           const float* __restrict__ dinv, float* __restrict__ agg, int E) {
    long j = (long)blockIdx.x * blockDim.x + threadIdx.x;
    if (j >= (long)E * D) return;
    int e = (int)(j / D);
    int c = (int)(j % D);
    int s = ei[e];
    int d = ei[E + e];
    float nrm = dinv[s] * dinv[d];
    atomicAdd(&agg[(long)d * D + c], feat[(long)s * D + c] * nrm);
}

// z = relu(agg + b2); fused per-graph mean-pool partial sums
__global__ void k_z_pool(const float* __restrict__ agg, const float* __restrict__ b2,
                         const int* __restrict__ batch, float* __restrict__ z,
                         float* __restrict__ pooled, float* __restrict__ cnt, int n) {
    long j = (long)blockIdx.x * blockDim.x + threadIdx.x;
    if (j >= (long)n * 128) return;
    int i = (int)(j >> 7), c = (int)(j & 127);
    float v = fmaxf(agg[j] + b2[c], 0.0f);
    z[j] = v;
    int g = batch[i];
    atomicAdd(&pooled[(g << 7) + c], v);
    if (c == 0) atomicAdd(&cnt[g], 1.0f);
}

// ---------------------------------------------------------------------------
// WMMA f32 GEMM:  C[M,N] = act(A[M,K] @ B[K,N] + bias)
// Block: 256 threads = 8 waves, tile 64(M) x 32(N); each wave one 16x16 tile.
// K staged through LDS in chunks of 64 with async global->LDS copies.
// OOB A rows are clamped (not zeroed): A rows map 1:1 to C rows, and OOB
// rows are never stored, so clamped data cannot contaminate valid outputs.
// ---------------------------------------------------------------------------

#define KC 64

template<int K, int N, bool RELU, bool BIAS>
__global__ __launch_bounds__(256)
void wmma_gemm(const float* __restrict__ A, const float* __restrict__ B,
               const float* __restrict__ bias, float* __restrict__ C, int M) {
    __shared__ float As[64][KC + 4];   // pad -> conflict-free column reads
    __shared__ float Bs[KC][32];
    const int tid  = threadIdx.x;
    const int wave = tid >> 5, lane = tid & 31;
    const int wm = wave & 3, wn = wave >> 2;       // 4 (M) x 2 (N) waves
    const int half = lane >> 4, mr = lane & 15;
    const int m0 = blockIdx.y * 64;
    const int n0 = blockIdx.x * 32;

    v8f acc = {};
    for (int kc = 0; kc < K; kc += KC) {
        // ---- stage A: 64 x KC floats = 1024 b128 async copies, 4/thread ----
        for (int i = 0; i < 4; ++i) {
            int idx = tid + i * 256;               // float4 index
            int r = idx / (KC / 4), c4 = idx % (KC / 4);
            int gr = m0 + r;
            if (gr >= M) gr = M - 1;               // clamp (see note above)
            async_ld_b128(A + (long)gr * K + kc + c4 * 4, &As[r][c4 * 4]);
        }
        // ---- stage B: KC x 32 floats = 512 b128 async copies, 2/thread ----
        for (int i = 0; i < 2; ++i) {
            int idx = tid + i * 256;
            int r = idx / 8, c4 = idx % 8;
            async_ld_b128(B + (long)(kc + r) * N + n0 + c4 * 4, &Bs[r][c4 * 4]);
        }
        wait_async0();
        __syncthreads();
        // ---- 16 WMMAs over this K chunk ----
        for (int kk = 0; kk < KC; kk += 4) {
            int ka = kk + 2 * half;
            v2f a, b;
            a.x = As[wm * 16 + mr][ka];
            a.y = As[wm * 16 + mr][ka + 1];
            b.x = Bs[ka][wn * 16 + mr];
            b.y = Bs[ka + 1][wn * 16 + mr];
            acc = __builtin_amdgcn_wmma_f32_16x16x4_f32(false, a, false, b,
                                                        (short)0, acc, false, false);
        }
        __syncthreads();
    }
    // ---- epilogue: C/D layout -> VGPR j holds row (j + 8*half), col = lane&15 ----
    int col = n0 + wn * 16 + mr;
    float bv = BIAS ? bias[col] : 0.0f;
    for (int j = 0; j < 8; ++j) {
        int row = m0 + wm * 16 + j + 8 * half;
        if (row < M) {
            float v = acc[j] + bv;
            if (RELU) v = fmaxf(v, 0.0f);
            C[(long)row * N + col] = v;
        }
    }
}

// ---------------------------------------------------------------------------
// Fused GRU kernel (prev_h == 0):
//   xc = [z | x]  (two 128-wide K sources)
//   h  = (1 - sigmoid(xc@Wxz + bxz + bhz)) * tanh(xc@Wxh + bxh + bhh)
// Two B matrices, two accumulators per wave. N = 256, K = 256.
// ---------------------------------------------------------------------------

__global__ __launch_bounds__(256)
void gru_kernel(const float* __restrict__ Z, const float* __restrict__ X,
                const float* __restrict__ Wxz, const float* __restrict__ Wxh,
                const float* __restrict__ bxz, const float* __restrict__ bhz,
                const float* __restrict__ bxh, const float* __restrict__ bhh,
                float* __restrict__ H, int M) {
    __shared__ float As[64][KC + 4];
    __shared__ float Bzs[KC][32];
    __shared__ float Bhs[KC][32];
    const int tid  = threadIdx.x;
    const int wave = tid >> 5, lane = tid & 31;
    const int wm = wave & 3, wn = wave >> 2;
    const int half = lane >> 4, mr = lane & 15;
    const int m0 = blockIdx.y * 64;
    const int n0 = blockIdx.x * 32;

    v8f accz = {}, acch = {};
    for (int chunk = 0; chunk < 4; ++chunk) {          // K = 4 * 64
        const float* Asrc = (chunk < 2) ? Z : X;       // concat(z, x)
        const int kloc = (chunk & 1) * KC;             // offset within 128-wide source
        for (int i = 0; i < 4; ++i) {
            int idx = tid + i * 256;
            int r = idx / (KC / 4), c4 = idx % (KC / 4);
            int gr = m0 + r;
            if (gr >= M) gr = M - 1;
            async_ld_b128(Asrc + (long)gr * 128 + kloc + c4 * 4, &As[r][c4 * 4]);
        }
        for (int i = 0; i < 2; ++i) {
            int idx = tid + i * 256;
            int r = idx / 8, c4 = idx % 8;
            long brow = (long)(chunk * KC + r) * 256 + n0 + c4 * 4;
            async_ld_b128(Wxz + brow, &Bzs[r][c4 * 4]);
            async_ld_b128(Wxh + brow, &Bhs[r][c4 * 4]);
        }
        wait_async0();
        __syncthreads();
        for (int kk = 0; kk < KC; kk += 4) {
            int ka = kk + 2 * half;
            v2f a, bz, bh;
            a.x  = As[wm * 16 + mr][ka];
            a.y  = As[wm * 16 + mr][ka + 1];
            bz.x = Bzs[ka][wn * 16 + mr];
            bz.y = Bzs[ka + 1][wn * 16 + mr];
            bh.x = Bhs[ka][wn * 16 + mr];
            bh.y = Bhs[ka + 1][wn * 16 + mr];
            accz = __builtin_amdgcn_wmma_f32_16x16x4_f32(false, a, false, bz,
                                                         (short)0, accz, false, false);
            acch = __builtin_amdgcn_wmma_f32_16x16x4_f32(false, a, false, bh,
                                                         (short)0, acch, false, false);
        }
        __syncthreads();
    }
    int col = n0 + wn * 16 + mr;
    float bz = bxz[col] + bhz[col];
    float bh = bxh[col] + bhh[col];
    for (int j = 0; j < 8; ++j) {
        int row = m0 + wm * 16 + j + 8 * half;
        if (row < M) {
            float zg = 1.0f / (1.0f + expf(-(accz[j] + bz)));
            float ht = tanhf(acch[j] + bh);
            H[(long)row * 256 + col] = (1.0f - zg) * ht;
        }
    }
}

// ---------------------------------------------------------------------------
// Head: graph_emb = pooled/cnt ; global_emb = relu(u@Wg + bg) ;
//       out = [graph_emb | global_emb] @ Whead + bhead      (single block)
// ---------------------------------------------------------------------------

__global__ __launch_bounds__(256)
void head_kernel(const float* __restrict__ u, const float* __restrict__ Wg,
                 const float* __restrict__ bg, const float* __restrict__ pooled,
                 const float* __restrict__ cnt, const float* __restrict__ Whead,
                 const float* __restrict__ bhead, float* __restrict__ out) {
    __shared__ float gemb[NG * 128];
    for (int idx = threadIdx.x; idx < NG * 128; idx += 256) {
        int g = idx >> 7, c = idx & 127;
        float acc = bg[c];
        for (int k = 0; k < 32; ++k) acc += u[g * 32 + k] * Wg[k * 128 + c];
        gemb[idx] = fmaxf(acc, 0.0f);
    }
    __syncthreads();
    if (threadIdx.x < NG) {
        int g = threadIdx.x;
        float inv = 1.0f / fmaxf(cnt[g], 1.0f);
        float acc = bhead[0];
        for (int c = 0; c < 128; ++c) acc += pooled[g * 128 + c] * inv * Whead[c];
        for (int c = 0; c < 128; ++c) acc += gemb[g * 128 + c] * Whead[128 + c];
        out[g] = acc;
    }
}

// ---------------------------------------------------------------------------
// Launch
// ---------------------------------------------------------------------------

extern "C" void kernel_launch(void* const* d_in, const int* in_sizes, int n_in,
                              void* d_out, int out_size, void* d_ws, size_t ws_size,
                              hipStream_t stream) {
    const float* x     = (const float*)d_in[0];
    const float* u     = (const float*)d_in[1];
    const int*   ei    = (const int*)d_in[2];
    const int*   batch = (const int*)d_in[3];
    const float* W1    = (const float*)d_in[5];
    const float* b1    = (const float*)d_in[6];
    const float* W2    = (const float*)d_in[7];
    const float* b2    = (const float*)d_in[8];
    const float* Wg    = (const float*)d_in[9];
    const float* bg    = (const float*)d_in[10];
    const float* Wxz   = (const float*)d_in[11];
    const float* bxz   = (const float*)d_in[12];
    const float* bhz   = (const float*)d_in[14];
    const float* Wxh   = (const float*)d_in[19];
    const float* bxh   = (const float*)d_in[20];
    const float* bhh   = (const float*)d_in[22];
    const float* Whead = (const float*)d_in[23];
    const float* bhead = (const float*)d_in[24];

    float* outv = (float*)d_out;        // [0:64)  = out
    float* hout = outv + NG;            // [64:..) = h, row-major 50000x256

    // workspace layout
    float* ws     = (float*)d_ws;
    float* dinv   = ws;                            // NN          (deg -> dinv)
    float* B1     = dinv + NN;                     // NN*128      (aggX, then h2)
    float* B2     = B1 + (long)NN * 128;           // NN*256      (f, then z in first half)
    float* B3     = B2 + (long)NN * 256;           // NN*128      (aggH2)
    float* pooled = B3 + (long)NN * 128;           // 64*128
    float* cntb   = pooled + NG * 128;             // 64

    const int T = 256;
    const int nodeBlocks  = (NN + T - 1) / T;                    // 196
    const int edgeBlocks  = (NE + T - 1) / T;                    // 3125
    const int nd128Blocks = (int)(((long)NN * 128 + T - 1) / T); // 25000
    const int ed128Blocks = (int)(((long)NE * 128 + T - 1) / T); // 400000
    const dim3 gemmGridA(256 / 32, (NN + 63) / 64);              // N=256 -> 8 x 782
    const dim3 gemmGridB(128 / 32, (NN + 63) / 64);              // N=128 -> 4 x 782

    // degrees & normalization
    k_init<<<nodeBlocks, T, 0, stream>>>(dinv, pooled, cntb, NN);
    k_degree<<<edgeBlocks, T, 0, stream>>>(ei, dinv, NE);
    k_rsqrt<<<nodeBlocks, T, 0, stream>>>(dinv, NN);

    // layer 1: aggregate x first (128-dim), then GEMM  f = relu(aggX@W1 + b1)
    k_self<128><<<nd128Blocks, T, 0, stream>>>(x, dinv, B1, NN);
    k_edge_agg<128><<<ed128Blocks, T, 0, stream>>>(x, ei, dinv, B1, NE);
    wmma_gemm<128, 256, true, true><<<gemmGridA, T, 0, stream>>>(B1, W1, b1, B2, NN);

    // layer 2: GEMM first (256->128), then aggregate;  z = relu(agg + b2)
    wmma_gemm<256, 128, false, false><<<gemmGridB, T, 0, stream>>>(B2, W2, nullptr, B1, NN);
    k_self<128><<<nd128Blocks, T, 0, stream>>>(B1, dinv, B3, NN);
    k_edge_agg<128><<<ed128Blocks, T, 0, stream>>>(B1, ei, dinv, B3, NE);
    k_z_pool<<<nd128Blocks, T, 0, stream>>>(B3, b2, batch, B2, pooled, cntb, NN);

    // GRU (prev_h = 0): h = (1 - sigmoid(xc@Wxz+bxz+bhz)) * tanh(xc@Wxh+bxh+bhh)
    gru_kernel<<<gemmGridA, T, 0, stream>>>(B2, x, Wxz, Wxh, bxz, bhz, bxh, bhh, hout, NN);

    // pooling head -> out[0:64]
    head_kernel<<<1, T, 0, stream>>>(u, Wg, bg, pooled, cntb, Whead, bhead, outv);
}